// StructuredMeanField_32031866093592
// MI455X (gfx1250) — compile-verified
//
#include <hip/hip_runtime.h>
#include <math.h>

typedef float v2f __attribute__((ext_vector_type(2)));
typedef float v8f __attribute__((ext_vector_type(8)));
typedef unsigned int v4u __attribute__((ext_vector_type(4)));
typedef int v4i __attribute__((ext_vector_type(4)));
typedef int v8i __attribute__((ext_vector_type(8)));

#if defined(__has_builtin)
#if __has_builtin(__builtin_amdgcn_tensor_load_to_lds)
#define HAS_TDM 1
#endif
#endif
#ifndef HAS_TDM
#define HAS_TDM 0
#endif

#define EPSV 1e-6f

__device__ __forceinline__ float sigmoid_clip(float x) {
    float s = 1.0f / (1.0f + __expf(-x));
    return fminf(fmaxf(s, EPSV), 1.0f - EPSV);
}

__device__ __forceinline__ float bin_ent(float q) {
    return -(q * __logf(q) + (1.0f - q) * __logf(1.0f - q));
}

// ---------------- TDM-staged scan kernels (n == 4096 geometry) ----------------
// Block = 128 threads = 2 rows x 64 segments; each thread owns one 64-column
// segment. Columns are processed in 4 chunks of 16; each chunk of the whole
// block (128 lines x 32 dwords, global stride 128 dwords) is DMA'd to LDS by
// the Tensor Data Mover with +2-dword padding every 32 dwords so per-thread
// sequential reads are bank-conflict-free (line stride 34 dwords, 8B aligned).

#define TD_ROWS 2
#define TD_SEG  64
#define TD_L    64
#define TD_CH   16
#define TD_NQ   (TD_L / TD_CH)
#define TD_LINES (TD_ROWS * TD_SEG)   /* 128 threads per block */
#define TD_LDW  34                    /* 32 data dwords + 2 pad dwords */

__device__ __forceinline__ void tdm_issue_tile(const float* gline0, const float* lds_dst) {
#if HAS_TDM
    unsigned long long ga = (unsigned long long)(uintptr_t)gline0;
    unsigned int lo = (unsigned int)(uintptr_t)lds_dst;     // LDS byte offset
    v4u g0;
    g0.x = 1u;                                              // count=1 (valid user D#)
    g0.y = lo;                                              // lds_addr
    g0.z = (unsigned int)(ga & 0xffffffffull);              // global_addr[31:0]
    g0.w = (unsigned int)((ga >> 32) & 0x01ffffffull) | 0x80000000u; // [56:32] | type=2
    v8i g1;
    g1[0] = (2 << 16)      /* data_size = 4B */
          | (1 << 20)      /* pad_enable */
          | (4 << 22)      /* pad_interval: every 32 dwords */
          | (1 << 25);     /* pad_amount: 2 dwords */
    g1[1] = (TD_CH * 2) << 16;   // tensor_dim0 = 32 elements (low 16 bits)
    g1[2] = TD_LINES << 16;      // tensor_dim1 = 128 (low 16 bits)
    g1[3] = (TD_CH * 2) << 16;   // tile_dim0 = 32
    g1[4] = TD_LINES;            // tile_dim1 = 128 (tile_dim2 = 0)
    g1[5] = TD_L * 2;            // tensor_dim0_stride = 128 elements
    g1[6] = 0;
    g1[7] = 0;
    v4i z4 = {0, 0, 0, 0};
#if __clang_major__ >= 23
    v8i z8 = {0, 0, 0, 0, 0, 0, 0, 0};
    __builtin_amdgcn_tensor_load_to_lds(g0, g1, z4, z4, z8, 0);
#else
    __builtin_amdgcn_tensor_load_to_lds(g0, g1, z4, z4, 0);
#endif
#else
    (void)gline0; (void)lds_dst;
#endif
}

#if !HAS_TDM
__device__ __forceinline__ void stage_fallback(const float2* gline, float* ldsline) {
    for (int k = 0; k < TD_CH; ++k) ((float2*)ldsline)[k] = gline[k];
}
#endif

__global__ void __launch_bounds__(TD_LINES)
k_compose_tdm(const float* __restrict__ param,
              float* __restrict__ segA, float* __restrict__ segB,
              float* __restrict__ segc, float* __restrict__ segd, int n) {
    __shared__ __align__(16) float lds[2][TD_LINES * TD_LDW];
    int t = threadIdx.x;
    const float* gblk = param + (size_t)blockIdx.x * TD_ROWS * n * 2;
    bool issuer = (t >> 5) == 0;   // wave 0 (uniform per wave)
#if HAS_TDM
    if (issuer) tdm_issue_tile(gblk, &lds[0][0]);
#else
    stage_fallback((const float2*)gblk + (size_t)t * TD_L, &lds[0][t * TD_LDW]);
#endif
    float A = 1.0f, B = 0.0f, c = 0.0f, d = 0.0f;
    for (int q = 0; q < TD_NQ; ++q) {
        int cur = q & 1, nxt = cur ^ 1;
#if HAS_TDM
        if (issuer) {
            if (q + 1 < TD_NQ) {
                tdm_issue_tile(gblk + (q + 1) * TD_CH * 2, &lds[nxt][0]);
                __builtin_amdgcn_s_wait_tensorcnt(1);   // oldest DMA (chunk q) done
            } else {
                __builtin_amdgcn_s_wait_tensorcnt(0);
            }
        }
#else
        if (q + 1 < TD_NQ)
            stage_fallback((const float2*)gblk + (size_t)t * TD_L + (q + 1) * TD_CH,
                           &lds[nxt][t * TD_LDW]);
#endif
        __syncthreads();
        const float2* line = (const float2*)&lds[cur][t * TD_LDW];
        for (int jj2 = 0; jj2 < TD_CH; ++jj2) {
            float2 v = line[jj2];
            float p0 = sigmoid_clip(v.x);
            float p1 = sigmoid_clip(v.y);
            if ((t % TD_SEG) == 0 && q == 0 && jj2 == 0) p1 = p0; // column 0 of a row
            float H0 = bin_ent(p0);
            float dH = bin_ent(p1) - H0;
            c = fmaf(B, dH, c + H0);
            d = fmaf(A, dH, d);
            float m = p1 - p0;
            B = fmaf(B, m, p0);
            A *= m;
        }
        __syncthreads();   // buffer free before it is re-issued
    }
    int tid = blockIdx.x * TD_LINES + t;
    segA[tid] = A; segB[tid] = B; segc[tid] = c; segd[tid] = d;
}

__global__ void __launch_bounds__(TD_LINES)
k_means_tdm(const float* __restrict__ param, const float* __restrict__ segP,
            float* __restrict__ mean, int n) {
    __shared__ __align__(16) float lds[2][TD_LINES * TD_LDW];
    int t = threadIdx.x;
    const float* gblk = param + (size_t)blockIdx.x * TD_ROWS * n * 2;
    float* mblk = mean + (size_t)blockIdx.x * TD_ROWS * n;
    bool issuer = (t >> 5) == 0;
#if HAS_TDM
    if (issuer) tdm_issue_tile(gblk, &lds[0][0]);
#else
    stage_fallback((const float2*)gblk + (size_t)t * TD_L, &lds[0][t * TD_LDW]);
#endif
    float prev = segP[blockIdx.x * TD_LINES + t];
    for (int q = 0; q < TD_NQ; ++q) {
        int cur = q & 1, nxt = cur ^ 1;
#if HAS_TDM
        if (issuer) {
            if (q + 1 < TD_NQ) {
                tdm_issue_tile(gblk + (q + 1) * TD_CH * 2, &lds[nxt][0]);
                __builtin_amdgcn_s_wait_tensorcnt(1);
            } else {
                __builtin_amdgcn_s_wait_tensorcnt(0);
            }
        }
#else
        if (q + 1 < TD_NQ)
            stage_fallback((const float2*)gblk + (size_t)t * TD_L + (q + 1) * TD_CH,
                           &lds[nxt][t * TD_LDW]);
#endif
        __syncthreads();
        const float2* line = (const float2*)&lds[cur][t * TD_LDW];
        float out16[TD_CH];
        for (int jj2 = 0; jj2 < TD_CH; ++jj2) {
            float2 v = line[jj2];
            float p0 = sigmoid_clip(v.x);
            float p1 = sigmoid_clip(v.y);
            float curv = fmaf(prev, p1 - p0, p0);
            out16[jj2] = curv;
            prev = curv;
        }
        float* gm = mblk + (size_t)t * TD_L + q * TD_CH;   // 64B aligned
        for (int k = 0; k < 4; ++k) {
            float4 w; w.x = out16[4*k]; w.y = out16[4*k+1];
            w.z = out16[4*k+2]; w.w = out16[4*k+3];
            ((float4*)gm)[k] = w;
        }
        __syncthreads();
    }
}

// ---------------- generic (non-4096) fallback scan kernels ----------------
__global__ void k_compose(const float2* __restrict__ pp,
                          float* __restrict__ segA, float* __restrict__ segB,
                          float* __restrict__ segc, float* __restrict__ segd,
                          int n, int SEG, int L) {
    long tid = (long)blockIdx.x * blockDim.x + threadIdx.x;
    long total = (long)n * SEG;
    if (tid >= total) return;
    int row = (int)(tid / SEG);
    int seg = (int)(tid % SEG);
    long base = (long)row * n + (long)seg * L;
    float A = 1.0f, B = 0.0f, c = 0.0f, d = 0.0f;
    for (int jj = 0; jj < L; ++jj) {
        if ((jj & 15) == 0) __builtin_prefetch(pp + base + jj + 16, 0, 0);
        float2 q = pp[base + jj];
        float p0 = sigmoid_clip(q.x);
        float p1 = sigmoid_clip(q.y);
        if (seg == 0 && jj == 0) p1 = p0;
        float H0 = bin_ent(p0);
        float dH = bin_ent(p1) - H0;
        c = fmaf(B, dH, c + H0);
        d = fmaf(A, dH, d);
        float m = p1 - p0;
        B = fmaf(B, m, p0);
        A *= m;
    }
    segA[tid] = A; segB[tid] = B; segc[tid] = c; segd[tid] = d;
}

__global__ void k_means(const float2* __restrict__ pp, const float* __restrict__ segP,
                        float* __restrict__ mean, int n, int SEG, int L) {
    long tid = (long)blockIdx.x * blockDim.x + threadIdx.x;
    long total = (long)n * SEG;
    if (tid >= total) return;
    int row = (int)(tid / SEG);
    int seg = (int)(tid % SEG);
    long base = (long)row * n + (long)seg * L;
    float prev = segP[tid];
    for (int jj = 0; jj < L; ++jj) {
        if ((jj & 15) == 0) __builtin_prefetch(pp + base + jj + 16, 0, 0);
        float2 q = pp[base + jj];
        float p0 = sigmoid_clip(q.x);
        float p1 = sigmoid_clip(q.y);
        float cur = fmaf(prev, p1 - p0, p0);
        mean[base + jj] = cur;
        prev = cur;
    }
}

// Pass 2: per-row serial chain over SEG segments.
__global__ void k_prefix(const float* __restrict__ segA, const float* __restrict__ segB,
                         const float* __restrict__ segc, const float* __restrict__ segd,
                         float* __restrict__ segP, float* __restrict__ rowEnt,
                         int n, int SEG) {
    int row = blockIdx.x * blockDim.x + threadIdx.x;
    if (row >= n) return;
    long base = (long)row * SEG;
    float prev = 0.0f, ent = 0.0f;
    for (int s = 0; s < SEG; ++s) {
        segP[base + s] = prev;
        ent = fmaf(segd[base + s], prev, ent + segc[base + s]);
        prev = fmaf(segA[base + s], prev, segB[base + s]);
    }
    rowEnt[row] = ent;
}

// Final entropy reduction with WMMA (D = A x ones + C accumulates row-sums).
__global__ void k_entropy_wmma(const float* __restrict__ rowEnt, float* __restrict__ out, int n) {
    int lane = threadIdx.x;               // 32 threads, single wave, EXEC all-ones
    v8f acc = {0.f, 0.f, 0.f, 0.f, 0.f, 0.f, 0.f, 0.f};
    v2f ones; ones.x = 1.0f; ones.y = 1.0f;
    int iters = n / 64;
    const float2* p2 = (const float2*)rowEnt;
    for (int it = 0; it < iters; ++it) {
        float2 v = p2[it * 32 + lane];
        v2f a; a.x = v.x; a.y = v.y;
        acc = __builtin_amdgcn_wmma_f32_16x16x4_f32(
            false, a, false, ones, (short)0, acc, false, false);
    }
    float s = acc[0] + acc[1] + acc[2] + acc[3] + acc[4] + acc[5] + acc[6] + acc[7];
    float tot = __shfl(s, 0, 32) + __shfl(s, 16, 32);
    float tail = 0.0f;
    for (int i = iters * 64 + lane; i < n; i += 32) tail += rowEnt[i];
    for (int off = 16; off; off >>= 1) tail += __shfl_xor(tail, off, 32);
    if (lane == 0) out[0] = tot + tail;
}

__global__ void k_chain(const int2* __restrict__ ct, const float* __restrict__ mean,
                        const float2* __restrict__ pp, float* __restrict__ out, long T) {
    long i = (long)blockIdx.x * blockDim.x + threadIdx.x;
    if (i >= T) return;
    int2 t = ct[i];
    float ma = mean[t.x];
    float2 q = pp[t.y];
    float p0 = sigmoid_clip(q.x);
    float p1 = sigmoid_clip(q.y);
    float val = ma * p1 + (1.0f - ma) * (1.0f - p0);
    out[i] = 2.0f * val - 1.0f;
}

__global__ void k_hop(const int2* __restrict__ ht, const float* __restrict__ mean,
                      float* __restrict__ out, long T) {
    long i = (long)blockIdx.x * blockDim.x + threadIdx.x;
    if (i >= T) return;
    int2 t = ht[i];
    out[i] = (2.0f * mean[t.x] - 1.0f) * (2.0f * mean[t.y] - 1.0f);
}

extern "C" void kernel_launch(void* const* d_in, const int* in_sizes, int n_in,
                              void* d_out, int out_size, void* d_ws, size_t ws_size,
                              hipStream_t stream) {
    const float* param = (const float*)d_in[0];
    const int*   chain = (const int*)d_in[1];
    const int*   hop   = (const int*)d_in[2];

    long V = (long)in_sizes[0] / 2;
    int  n = (int)(sqrt((double)V) + 0.5);
    long T = (long)in_sizes[1] / 2;

    int SEG = 64;
    while (SEG > 1 && (n % SEG)) SEG >>= 1;
    int L = n / SEG;
    long nSegTot = (long)n * SEG;

    // workspace (floats): mean[V] | A | B | c | d | segP (nSegTot each) | rowEnt[n]
    float* ws     = (float*)d_ws;
    float* mean   = ws;
    float* segA   = ws + V;
    float* segB   = segA + nSegTot;
    float* segc   = segB + nSegTot;
    float* segd   = segc + nSegTot;
    float* segP   = segd + nSegTot;
    float* rowEnt = segP + nSegTot;

    bool tdm_geom = (SEG == TD_SEG) && (L == TD_L) && (n % TD_ROWS == 0);
    dim3 blk(256);
    if (tdm_geom) {
        int gb = n / TD_ROWS;    // 2048 blocks of 128 threads
        k_compose_tdm<<<gb, TD_LINES, 0, stream>>>(param, segA, segB, segc, segd, n);
    } else {
        int g1 = (int)((nSegTot + 255) / 256);
        k_compose<<<g1, blk, 0, stream>>>((const float2*)param, segA, segB, segc, segd, n, SEG, L);
    }
    k_prefix<<<(n + 255) / 256, blk, 0, stream>>>(segA, segB, segc, segd, segP, rowEnt, n, SEG);
    if (tdm_geom) {
        int gb = n / TD_ROWS;
        k_means_tdm<<<gb, TD_LINES, 0, stream>>>(param, segP, mean, n);
    } else {
        int g1 = (int)((nSegTot + 255) / 256);
        k_means<<<g1, blk, 0, stream>>>((const float2*)param, segP, mean, n, SEG, L);
    }
    k_entropy_wmma<<<1, 32, 0, stream>>>(rowEnt, (float*)d_out, n);

    int gT = (int)((T + 255) / 256);
    k_chain<<<gT, blk, 0, stream>>>((const int2*)chain, mean, (const float2*)param,
                                    (float*)d_out + 1, T);
    k_hop<<<gT, blk, 0, stream>>>((const int2*)hop, mean, (float*)d_out + 1 + T, T);
}